// CAR_CFL_Layer_48893907698291
// MI455X (gfx1250) — compile-verified
//
#include <hip/hip_runtime.h>

// ---------------------------------------------------------------------------
// CAR/CFL capsule routing layer for MI455X (gfx1250).
//   B=32, N_IN=2048, D_IN=16, N_OUT=32, CAP=32
// Roofline: W_ij 128MB + W_cf 276MB + inputs 4MB ~= 408MB @ 23.3TB/s ~= 17.5us
// floor. Math via V_WMMA_F32_16X16X4_F32 (2048 FLOP/instr) so f32 compute
// stays under the memory floor. Weight streaming uses the gfx1250 async
// global->LDS path (GLOBAL_LOAD_ASYNC_TO_LDS_B128, ASYNCcnt) to avoid the
// VGPR round-trip on the 408MB stream.
// ---------------------------------------------------------------------------

typedef __attribute__((ext_vector_type(2))) float v2f;
typedef __attribute__((ext_vector_type(8))) float v8f;

#define B_SZ  32
#define N_INC 2048
#define D_IN  16
#define N_OUT 32
#define CAP   32
#define KCF   33          // 1 + CAP
#define NB    8           // input capsules per workgroup
#define WGS   1024        // 32 wave32 waves; wave w <-> output capsule o=w
#define EPSQ  1e-7f

// ---- LDS partition (float offsets) ----------------------------------------
// Strides: 20 floats (inputs / W_ij rows) -> bank-conflict-free (gcd(20,64)=4
// gives 16 distinct banks for 16 lanes) AND 16B-aligned rows for b128 async.
// 33 floats (p_j / W_cf rows) -> odd stride, conflict-free.
#define IN_STR   20
#define P_STR    33
#define WST_STR  1088                     // per-wave staging: >= 32*20 and 32*33
#define OFF_IN   0
#define OFF_BIJ  (OFF_IN  + B_SZ*IN_STR)              // 640
#define OFF_CIJ  (OFF_BIJ + B_SZ*N_OUT)               // +1024
#define OFF_P    (OFF_CIJ + N_OUT*B_SZ)               // +1024
#define OFF_WST  (OFF_P   + N_OUT*B_SZ*P_STR)         // +33792
#define SMEM_FLOATS (OFF_WST + N_OUT*WST_STR)         // 71296 fl = 285184 B

// ---- gfx1250 async global->LDS copies (ASYNCcnt-tracked) ------------------
__device__ __forceinline__ unsigned int lds_addr(const void* p) {
  // ISA 10.2: generic->LDS aperture mapping uses addr[31:0] as the LDS address
  return (unsigned int)(unsigned long long)p;
}
__device__ __forceinline__ void async_copy_b128(const void* g, const void* l) {
  asm volatile("global_load_async_to_lds_b128 %0, %1, off"
               :: "v"(lds_addr(l)), "v"((unsigned long long)g) : "memory");
}
__device__ __forceinline__ void async_copy_b32(const void* g, const void* l) {
  asm volatile("global_load_async_to_lds_b32 %0, %1, off"
               :: "v"(lds_addr(l)), "v"((unsigned long long)g) : "memory");
}
__device__ __forceinline__ void wait_async0() {
  asm volatile("s_wait_asynccnt 0x0" ::: "memory");
}

__global__ void __launch_bounds__(256) car_zero(float* w, int nfl) {
  int i = blockIdx.x * blockDim.x + threadIdx.x;
  if (i < nfl) w[i] = 0.0f;
}

__global__ void __launch_bounds__(WGS, 1)
car_main(const float* __restrict__ inp,   // [B, N_IN, D_IN]
         const float* __restrict__ Wij,   // [N_IN, N_OUT, CAP, D_IN]
         const float* __restrict__ Wcf,   // [N_IN, N_OUT, CAP, 1+CAP]
         float* __restrict__ wsacc) {     // [B, N_OUT, 48] f32 accumulator
  extern __shared__ float smem[];
  const int tid  = threadIdx.x;
  const int lane = tid & 31;
  const int o    = tid >> 5;          // wave index == output capsule
  const int ln   = lane & 15;         // lane within half-wave
  const int hi   = lane >> 4;         // 0: lanes 0-15, 1: lanes 16-31
  const int kh   = hi * 2;            // WMMA A/B K offset for upper half

  float* in_lds = smem + OFF_IN;           // [b][IN_STR]
  float* bij    = smem + OFF_BIJ;          // [b][o]
  float* cij    = smem + OFF_CIJ;          // [o][b]
  float* plds   = smem + OFF_P;            // [o][b][P_STR]
  float* wst    = smem + OFF_WST + o * WST_STR;  // per-wave W staging

  // Running s_j partials in registers, in WMMA C/D tile layout:
  // element (b = mt*16 + r + hi*8, col), col = ln (inputs) or nt*16+ln (F part)
  float sIn[2][8];
  float sF[2][2][8];
#pragma unroll
  for (int mt = 0; mt < 2; ++mt)
#pragma unroll
    for (int r = 0; r < 8; ++r) {
      sIn[mt][r] = 0.0f;
      sF[mt][0][r] = 0.0f;
      sF[mt][1][r] = 0.0f;
    }

  const int n0 = blockIdx.x * NB;
  for (int nn = 0; nn < NB; ++nn) {
    const int n = n0 + nn;

    // ---- phase 1: async-stage inputs[:,n,:] and W_ij[n,o] into LDS --------
    if (o == 0) {                          // wave 0: inputs, 4x b128 per lane
      const float* src = inp + ((size_t)lane * N_INC + n) * D_IN;
#pragma unroll
      for (int j = 0; j < 4; ++j)
        async_copy_b128(src + j * 4, in_lds + lane * IN_STR + j * 4);
    }
    {
      const float* src = Wij + ((size_t)n * N_OUT + o) * (CAP * D_IN);
#pragma unroll
      for (int j = 0; j < 4; ++j) {        // 512 f32 = 128 vec4, all lanes
        int v = j * 32 + lane;             // vec4 id; row = v/4 (16 f32/row)
        async_copy_b128(src + v * 4, wst + (v >> 2) * IN_STR + (v & 3) * 4);
      }
    }
    wait_async0();
    __syncthreads();

    // ---- phase 2: GEMM1  p_o[32b x 32c] = inputs[32x16] * Wij[n,o]^T ------
    v8f acc[2][2] = {};
#pragma unroll
    for (int ks = 0; ks < 4; ++ks) {
      const int k0 = ks * 4 + kh;
      v2f a[2], bm[2];
#pragma unroll
      for (int mt = 0; mt < 2; ++mt) {
        a[mt].x = in_lds[(mt * 16 + ln) * IN_STR + k0];
        a[mt].y = in_lds[(mt * 16 + ln) * IN_STR + k0 + 1];
      }
#pragma unroll
      for (int nt = 0; nt < 2; ++nt) {
        bm[nt].x = wst[(nt * 16 + ln) * IN_STR + k0];
        bm[nt].y = wst[(nt * 16 + ln) * IN_STR + k0 + 1];
      }
#pragma unroll
      for (int mt = 0; mt < 2; ++mt)
#pragma unroll
        for (int nt = 0; nt < 2; ++nt)
          acc[mt][nt] = __builtin_amdgcn_wmma_f32_16x16x4_f32(
              false, a[mt], false, bm[nt], (short)0, acc[mt][nt], false, false);
    }

    // norms (tf.norm over c) + park p_j in LDS for GEMM2 A-operand reload
#pragma unroll
    for (int mt = 0; mt < 2; ++mt) {
#pragma unroll
      for (int r = 0; r < 8; ++r) {
        float t0 = acc[mt][0][r], t1 = acc[mt][1][r];
        float v = t0 * t0 + t1 * t1;
        v += __shfl_xor(v, 1, 32);
        v += __shfl_xor(v, 2, 32);
        v += __shfl_xor(v, 4, 32);
        v += __shfl_xor(v, 8, 32);         // reduce within each half-wave
        int b = mt * 16 + r + hi * 8;
        if (ln == 0) bij[b * N_OUT + o] = sqrtf(v);
        plds[o * (B_SZ * P_STR) + b * P_STR + ln]      = t0;
        plds[o * (B_SZ * P_STR) + b * P_STR + 16 + ln] = t1;
      }
    }
    __syncthreads();                       // W_ij staging dead after this

    // ---- phase 3: softmax over o (wave 0) + async-stage W_cf[n,o] ---------
    {
      const float* src = Wcf + ((size_t)n * N_OUT + o) * (CAP * KCF);
#pragma unroll
      for (int j = 0; j < 8; ++j) {        // 1024 f32 = 256 vec4, all lanes
        int v = j * 32 + lane;
        async_copy_b128(src + (size_t)v * 4, wst + v * 4);  // flat [c][33]
      }
      async_copy_b32(src + 1024 + lane, wst + 1024 + lane); // 32-f32 tail
    }
    if (tid < B_SZ) {
      int b = tid;
      float m = -3.0e38f;
      for (int oo = 0; oo < N_OUT; ++oo) m = fmaxf(m, bij[b * N_OUT + oo]);
      float s = 0.0f;
      for (int oo = 0; oo < N_OUT; ++oo) s += __expf(bij[b * N_OUT + oo] - m);
      float invs = 1.0f / s;
      for (int oo = 0; oo < N_OUT; ++oo)
        cij[oo * B_SZ + b] = __expf(bij[b * N_OUT + oo] - m) * invs;
    }
    wait_async0();
    __syncthreads();

    // ---- phase 4: GEMM2  F_o[32b x 32c] = p_o[32x32] * Wcf[:,1:33]^T ------
    v8f acc2[2][2] = {};
    const float* po = plds + o * (B_SZ * P_STR);
#pragma unroll
    for (int ks = 0; ks < 8; ++ks) {
      const int k0 = ks * 4 + kh;
      v2f a[2], bm[2];
#pragma unroll
      for (int mt = 0; mt < 2; ++mt) {
        a[mt].x = po[(mt * 16 + ln) * P_STR + k0];
        a[mt].y = po[(mt * 16 + ln) * P_STR + k0 + 1];
      }
#pragma unroll
      for (int nt = 0; nt < 2; ++nt) {
        bm[nt].x = wst[(nt * 16 + ln) * KCF + 1 + k0];
        bm[nt].y = wst[(nt * 16 + ln) * KCF + 1 + k0 + 1];
      }
#pragma unroll
      for (int mt = 0; mt < 2; ++mt)
#pragma unroll
        for (int nt = 0; nt < 2; ++nt)
          acc2[mt][nt] = __builtin_amdgcn_wmma_f32_16x16x4_f32(
              false, a[mt], false, bm[nt], (short)0, acc2[mt][nt], false, false);
    }

    // rank-1 C_ij column of W_cf + weighted s_j accumulation
    float w0a = wst[ln * KCF];             // W_cf[n,o,c=ln,0]
    float w0b = wst[(16 + ln) * KCF];      // W_cf[n,o,c=16+ln,0]
#pragma unroll
    for (int mt = 0; mt < 2; ++mt) {
#pragma unroll
      for (int r = 0; r < 8; ++r) {
        int b = mt * 16 + r + hi * 8;
        float C = cij[o * B_SZ + b];       // broadcast within half-wave
        sIn[mt][r] += C * in_lds[b * IN_STR + ln];
        float F0 = acc2[mt][0][r] + C * w0a;
        float F1 = acc2[mt][1][r] + C * w0b;
        sF[mt][0][r] += C * F0;
        sF[mt][1][r] += C * F1;
      }
    }
    __syncthreads();                       // before next n overwrites staging
  }

  // ---- drain partials into the global [B, N_OUT, 48] accumulator ----------
#pragma unroll
  for (int mt = 0; mt < 2; ++mt) {
#pragma unroll
    for (int r = 0; r < 8; ++r) {
      int b = mt * 16 + r + hi * 8;
      float* dst = wsacc + ((size_t)b * N_OUT + o) * 48;
      atomicAdd(dst + ln,      sIn[mt][r]);
      atomicAdd(dst + 16 + ln, sF[mt][0][r]);
      atomicAdd(dst + 32 + ln, sF[mt][1][r]);
    }
  }
}

__global__ void __launch_bounds__(256)
car_squash(const float* __restrict__ wsacc, float* __restrict__ out) {
  int i = blockIdx.x * blockDim.x + threadIdx.x;   // (b,o) pair
  if (i >= B_SZ * N_OUT) return;
  const float* s = wsacc + (size_t)i * 48;
  float s2 = 0.0f;
#pragma unroll
  for (int j = 0; j < 48; ++j) { float v = s[j]; s2 += v * v; }
  float scale = s2 / (1.0f + s2) / sqrtf(s2 + EPSQ);
#pragma unroll
  for (int j = 0; j < 48; ++j) out[(size_t)i * 48 + j] = scale * s[j];
}

extern "C" void kernel_launch(void* const* d_in, const int* in_sizes, int n_in,
                              void* d_out, int out_size, void* d_ws, size_t ws_size,
                              hipStream_t stream) {
  (void)in_sizes; (void)n_in; (void)out_size; (void)ws_size;
  const float* inp = (const float*)d_in[0];
  const float* Wij = (const float*)d_in[1];
  const float* Wcf = (const float*)d_in[2];
  float* out   = (float*)d_out;
  float* wsacc = (float*)d_ws;                      // 192KB accumulator

  const int accN = B_SZ * N_OUT * 48;
  car_zero<<<dim3((accN + 255) / 256), dim3(256), 0, stream>>>(wsacc, accN);

  const size_t smem = (size_t)SMEM_FLOATS * sizeof(float);
  car_main<<<dim3(N_INC / NB), dim3(WGS), smem, stream>>>(inp, Wij, Wcf, wsacc);

  car_squash<<<dim3((B_SZ * N_OUT + 255) / 256), dim3(256), 0, stream>>>(wsacc, out);
}